// DotProductCredibility_55276229100112
// MI455X (gfx1250) — compile-verified
//
#include <hip/hip_runtime.h>

typedef __attribute__((ext_vector_type(2))) float v2f;
typedef __attribute__((ext_vector_type(8))) float v8f;

#define BQ 8
#define NQ 2048
#define NK 2048
#define ND 64
#define CHUNKS 8
#define KEYS_PER_CHUNK (NK / CHUNKS)   // 256 keys per block

// ---------------------------------------------------------------------------
// Kernel 1: masked partial column-sum of K.
// grid = (CHUNKS, B), block = 256.  Thread t: d = t&63, kk = t>>6.
// Each 64-thread group sweeps contiguous key rows -> fully coalesced b32 loads.
// partial[(b*CHUNKS + ch)*64 + d] = sum over this chunk's valid keys.
// ---------------------------------------------------------------------------
__global__ void ksum_partial_kernel(const float* __restrict__ keys,
                                    const int* __restrict__ valid_lens,
                                    float* __restrict__ partial) {
  const int b    = blockIdx.y;
  const int ch   = blockIdx.x;
  const int t    = threadIdx.x;      // 0..255
  const int d    = t & 63;
  const int kk   = t >> 6;           // 0..3
  const int vlen = valid_lens[b];

  const float* Kb   = keys + (size_t)b * NK * ND;
  const int    kbeg = ch * KEYS_PER_CHUNK + kk * (KEYS_PER_CHUNK / 4);

  float acc = 0.0f;
#pragma unroll 8
  for (int i = 0; i < KEYS_PER_CHUNK / 4; ++i) {
    const int k = kbeg + i;
    if (k < vlen) acc += Kb[(size_t)k * ND + d];
  }

  __shared__ float red[256];
  red[t] = acc;
  __syncthreads();
  if (t < 64) {
    partial[((size_t)b * CHUNKS + ch) * ND + t] =
        red[t] + red[t + 64] + red[t + 128] + red[t + 192];
  }
}

// ---------------------------------------------------------------------------
// Kernel 2: out[b,q] = (scale/(8*vl[b])) * dot(Q[b,q,:], Ksum[b,:])
// via V_WMMA_F32_16X16X4_F32, 16 queries per wave, K accumulated 4-at-a-time.
// grid = (NQ/128, B), block = 256 (8 wave32 -> 128 queries per block).
// ---------------------------------------------------------------------------
__global__ void credibility_wmma_kernel(const float* __restrict__ queries,
                                        const float* __restrict__ partial,
                                        const int* __restrict__ valid_lens,
                                        const float* __restrict__ scale,
                                        float* __restrict__ out) {
  __shared__ float sk[ND];
  __shared__ float s_coef;

  const int b     = blockIdx.y;
  const int qtile = blockIdx.x;
  const int t     = threadIdx.x;

  if (t < ND) {
    float s = 0.0f;
#pragma unroll
    for (int c = 0; c < CHUNKS; ++c)
      s += partial[((size_t)b * CHUNKS + c) * ND + t];
    sk[t] = s;
  }
  if (t == 0) {
    // scale * (1/sqrt(64)) * (1/vl) = scale / (8*vl)
    s_coef = scale[0] / (8.0f * (float)valid_lens[b]);
  }
  __syncthreads();

  const int wave = t >> 5;       // wave32: 8 waves per block
  const int lane = t & 31;
  const int half = lane >> 4;    // 0: lanes 0-15, 1: lanes 16-31
  const int m    = lane & 15;    // A-matrix row M
  const int q0   = qtile * 128 + wave * 16;

  const float* qrow = queries + ((size_t)b * NQ + q0 + m) * ND;

  // A 16x4 f32 layout: lanes 0-15 hold K={0,1} in V0/V1, lanes 16-31 K={2,3}.
  // B 4x16: same K-per-lane-group split; Ksum broadcast across all N columns.
  v8f c = {};
#pragma unroll
  for (int i = 0; i < ND / 4; ++i) {
    const int d0 = 4 * i + 2 * half;
    v2f a = *(const v2f*)(qrow + d0);   // 8B aligned: qrow 256B aligned, d0 even
    v2f bb;
    bb.x = sk[d0];
    bb.y = sk[d0 + 1];
    c = __builtin_amdgcn_wmma_f32_16x16x4_f32(false, a, false, bb,
                                              (short)0, c, false, false);
  }

  // C/D 16x16 f32: lane 0 VGPR r = (M=r, N=0); lane 16 VGPR r = (M=8+r, N=0).
  // All N columns identical (B broadcast), so lanes 0 and 16 hold all 16 rows.
  if (lane == 0 || lane == 16) {
    const float coef = s_coef;
    float* o = out + (size_t)b * NQ + q0 + 8 * half;
#pragma unroll
    for (int r = 0; r < 8; ++r) o[r] = coef * c[r];
  }
}

// ---------------------------------------------------------------------------
extern "C" void kernel_launch(void* const* d_in, const int* in_sizes, int n_in,
                              void* d_out, int out_size, void* d_ws, size_t ws_size,
                              hipStream_t stream) {
  const float* queries    = (const float*)d_in[0];  // [8,2048,64] f32
  const float* keys       = (const float*)d_in[1];  // [8,2048,64] f32
  const float* scale      = (const float*)d_in[2];  // scalar f32
  const int*   valid_lens = (const int*)d_in[3];    // [8] i32
  float*       out        = (float*)d_out;          // [8,2048] f32
  float*       partial    = (float*)d_ws;           // 8*8*64 f32 = 16 KB scratch

  ksum_partial_kernel<<<dim3(CHUNKS, BQ), 256, 0, stream>>>(keys, valid_lens, partial);
  credibility_wmma_kernel<<<dim3(NQ / 128, BQ), 256, 0, stream>>>(
      queries, partial, valid_lens, scale, out);
}